// TimeStepCell_37804302139960
// MI455X (gfx1250) — compile-verified
//
#include <hip/hip_runtime.h>
#include <stdint.h>

// ---------------------------------------------------------------------------
// PML split-field wave timestep, MI455X (gfx1250, wave32).
// Memory-bound (~2.3 flop/byte) -> optimize HBM traffic + L2 reuse, not WMMA.
// CDNA5 paths: nontemporal (th:) hints, Tensor Data Mover (tensor_load_to_lds
// + s_wait_tensorcnt) staging the P tile into LDS.
// ---------------------------------------------------------------------------

#define NSRC 16
#define NZ   1000
#define NX   1000
#define DT   0.001f
#define HH   10.0f
#define INVH (1.0f / HH)

typedef float    v4f __attribute__((ext_vector_type(4)));
typedef unsigned v4u __attribute__((ext_vector_type(4)));
typedef unsigned v8u __attribute__((ext_vector_type(8)));

// Output offsets (element counts, return order: Px_n, Pz_n, Ax_n, Az_n, P)
#define OFF_PX 0
#define OFF_PZ (NSRC * NZ * NX)                         // 16,000,000
#define OFF_AX (2 * NSRC * NZ * NX)                     // 32,000,000
#define N_AX   (NSRC * (NZ - 2) * (NX - 1))             // 15,952,032
#define OFF_AZ (OFF_AX + N_AX)                          // 47,952,032
#define N_AZ   (NSRC * (NZ - 1) * (NX - 2))             // 15,952,032
#define OFF_P  (OFF_AZ + N_AZ)                          // 63,904,064

// ---------------------------------------------------------------------------
// Kernel 1: pressure update. 3-D grid: blockIdx.y = z, blockIdx.z = s,
// thread = 4 consecutive x points (float4). No integer division; the z-border
// guard is a block-uniform scalar branch.
// ---------------------------------------------------------------------------
__global__ __launch_bounds__(256) void k_pressure(
    const float* __restrict__ Px, const float* __restrict__ Pz,
    const float* __restrict__ Ax, const float* __restrict__ Az,
    const float* __restrict__ src, const float* __restrict__ m2dt,
    const float* __restrict__ sx, const float* __restrict__ sz,
    float* __restrict__ oPx, float* __restrict__ oPz, float* __restrict__ oP)
{
    const int xvI = threadIdx.x;          // float4 index within the row
    if (xvI >= NX / 4) return;            // 250 of 256 threads active
    const int z  = blockIdx.y;            // uniform per block
    const int s  = blockIdx.z;            // uniform per block
    const int xv = xvI * 4;

    const size_t b2 = (size_t)z * NX + xv;               // 2-D coeff offset
    const size_t b3 = (size_t)s * NZ * NX + b2;          // 3-D field offset

    // Streamed, last-use inputs: non-temporal b128 loads
    const v4f px4 = __builtin_nontemporal_load((const v4f*)(Px  + b3));
    const v4f pz4 = __builtin_nontemporal_load((const v4f*)(Pz  + b3));
    const v4f sc4 = __builtin_nontemporal_load((const v4f*)(src + b3));
    // 2-D coefficients: reused by all 16 shots -> regular (cacheable) loads
    const v4f m4  = *(const v4f*)(m2dt + b2);
    const v4f sx4 = *(const v4f*)(sx + b2);
    const v4f sz4 = *(const v4f*)(sz + b2);

    // Staggered derivative terms (zero on the 1-wide border ring).
    // z-guard is scalar (uniform); only lane 0 / lane 249 have masked x.
    float axx[4] = {0.f, 0.f, 0.f, 0.f};
    float azz[4] = {0.f, 0.f, 0.f, 0.f};
    if (z >= 1 && z <= NZ - 2) {
        const float* AxR  = Ax + (size_t)s * (NZ - 2) * (NX - 1)
                               + (size_t)(z - 1) * (NX - 1);
        const float* AzR1 = Az + (size_t)s * (NZ - 1) * (NX - 2)
                               + (size_t)z * (NX - 2);
        const float* AzR0 = AzR1 - (NX - 2);
#pragma unroll
        for (int i = 0; i < 4; ++i) {
            const int x = xv + i;
            if (x >= 1 && x <= NX - 2) {
                axx[i] = (AxR[x]      - AxR[x - 1])  * INVH;
                azz[i] = (AzR1[x - 1] - AzR0[x - 1]) * INVH;
            }
        }
    }

    v4f rPx, rPz, rP;
#pragma unroll
    for (int i = 0; i < 4; ++i) {
        const float ax  = sx4[i] * (0.5f * DT);
        const float az  = sz4[i] * (0.5f * DT);
        const float pxn = (m4[i] * axx[i] + (1.0f - ax) * px4[i]) / (1.0f + ax);
        const float pzn = (m4[i] * azz[i] + (1.0f - az) * pz4[i]) / (1.0f + az);
        rPx[i] = pxn;
        rPz[i] = pzn;
        rP[i]  = pxn + pzn + sc4[i] * DT;
    }

    __builtin_nontemporal_store(rPx, (v4f*)(oPx + b3));   // th:NT, never re-read
    __builtin_nontemporal_store(rPz, (v4f*)(oPz + b3));   // th:NT, never re-read
    *(v4f*)(oP + b3) = rP;                                // RT: kernel 2 reads via L2
}

// ---------------------------------------------------------------------------
// Kernel 2: velocity (Ax/Az) update. Stage a (BZ+1)x(BX+1) P tile into LDS
// with one Tensor Data Mover descriptor (2-D strided DMA), then both updates
// read the shared tile. Each P point feeds up to 4 stencil taps.
// ---------------------------------------------------------------------------
#define BZ 16
#define BX 64
#define TW (BX + 1)                 // 65 floats per tile row
#define TH_ROWS (BZ + 1)            // 17 tile rows
#define TILE_N (TH_ROWS * TW)       // 1105 floats

__global__ __launch_bounds__(256) void k_velocity(
    const float* __restrict__ P,
    const float* __restrict__ Ax, const float* __restrict__ Az,
    const float* __restrict__ sxh, const float* __restrict__ szh,
    float* __restrict__ oAx, float* __restrict__ oAz)
{
    __shared__ float tile[TILE_N];

    const int s  = blockIdx.z;
    const int z0 = blockIdx.y * BZ;
    const int x0 = blockIdx.x * BX;

    // --- TDM: DMA P[z0 .. z0+BZ][x0 .. x0+BX] into LDS ----------------------
    // One wave programs the descriptor and issues the tensor load; the block
    // barrier publishes the tile. tensor_dim0/1 encode the remaining extent
    // from the tile start so out-of-grid reads return zero (those cells only
    // feed masked outputs).
    if (threadIdx.x < 32) {
        const unsigned lds_addr = (unsigned)(uintptr_t)(&tile[0]);
        const unsigned long long gaddr = (unsigned long long)(uintptr_t)
            (P + ((size_t)s * NZ * NX + (size_t)z0 * NX + x0));
        const unsigned dim0 = (unsigned)(NX - x0);   // remaining cols
        const unsigned dim1 = (unsigned)(NZ - z0);   // remaining rows

        v4u g0;
        g0[0] = 1u;                                   // count=1 (valid D#)
        g0[1] = lds_addr;                             // LDS byte address
        g0[2] = (unsigned)(gaddr & 0xFFFFFFFFull);    // global_addr[31:0]
        g0[3] = ((unsigned)(gaddr >> 32) & 0x01FFFFFFu)
              | 0x80000000u;                          // addr[56:32] | type=2

        v8u g1;
        g1[0] = 0x00020000u;                          // data_size=2 (4 bytes)
        g1[1] = (dim0 & 0xFFFFu) << 16;               // tensor_dim0[15:0]
        g1[2] = (dim0 >> 16) | ((dim1 & 0xFFFFu) << 16);
        g1[3] = (dim1 >> 16) | ((unsigned)TW << 16);  // tile_dim0 = 65
        g1[4] = (unsigned)TH_ROWS;                    // tile_dim1 = 17
        g1[5] = (unsigned)NX;                         // tensor_dim0_stride lo
        g1[6] = 0u;                                   // stride hi / dim1 stride
        g1[7] = 0u;

        asm volatile("tensor_load_to_lds %0, %1" :: "s"(g0), "s"(g1) : "memory");
        __builtin_amdgcn_s_wait_tensorcnt(0);
    }
    __syncthreads();

    // --- compute: 4 rows per thread -----------------------------------------
    const int col = threadIdx.x % BX;        // 0..63 (coalesced per wave32)
    const int r4  = (threadIdx.x / BX) * 4;  // 0,4,8,12
    const float dt2 = 0.5f * DT;

#pragma unroll
    for (int j = 0; j < 4; ++j) {
        const int rr = r4 + j;
        const int zd = z0 + rr;
        const int xd = x0 + col;
        const float pc = tile[(rr + 1) * TW + col + 1];  // P[zd+1][xd+1]
        const float pl = tile[(rr + 1) * TW + col];      // P[zd+1][xd]
        const float pu = tile[rr * TW + col + 1];        // P[zd][xd+1]

        if (zd < NZ - 2 && xd < NX - 1) {                // Ax_n: (998, 999)
            const size_t ia = (size_t)s * (NZ - 2) * (NX - 1)
                            + (size_t)zd * (NX - 1) + xd;
            const float a  = __builtin_nontemporal_load(Ax + ia);   // last use
            const float sg = sxh[(size_t)zd * (NX - 1) + xd] * dt2; // reused x16
            const float px = (pc - pl) * INVH;
            __builtin_nontemporal_store(
                ((1.0f - sg) * a + DT * px) / (1.0f + sg), oAx + ia);
        }
        if (zd < NZ - 1 && xd < NX - 2) {                // Az_n: (999, 998)
            const size_t ib = (size_t)s * (NZ - 1) * (NX - 2)
                            + (size_t)zd * (NX - 2) + xd;
            const float a  = __builtin_nontemporal_load(Az + ib);
            const float sg = szh[(size_t)zd * (NX - 2) + xd] * dt2;
            const float pz = (pc - pu) * INVH;
            __builtin_nontemporal_store(
                ((1.0f - sg) * a + DT * pz) / (1.0f + sg), oAz + ib);
        }
    }
}

// ---------------------------------------------------------------------------
extern "C" void kernel_launch(void* const* d_in, const int* in_sizes, int n_in,
                              void* d_out, int out_size, void* d_ws, size_t ws_size,
                              hipStream_t stream)
{
    const float* Px   = (const float*)d_in[0];
    const float* Pz   = (const float*)d_in[1];
    const float* Ax   = (const float*)d_in[2];
    const float* Az   = (const float*)d_in[3];
    const float* src  = (const float*)d_in[4];
    const float* m2dt = (const float*)d_in[5];
    const float* sx   = (const float*)d_in[6];
    const float* sxh  = (const float*)d_in[7];
    const float* sz   = (const float*)d_in[8];
    const float* szh  = (const float*)d_in[9];

    float* out = (float*)d_out;
    float* oPx = out + OFF_PX;
    float* oPz = out + OFF_PZ;
    float* oAx = out + OFF_AX;
    float* oAz = out + OFF_AZ;
    float* oP  = out + OFF_P;

    // Pass 1: pressure. One block per (row, shot); 250 float4 threads per row.
    dim3 g1(1, NZ, NSRC);
    k_pressure<<<g1, 256, 0, stream>>>(
        Px, Pz, Ax, Az, src, m2dt, sx, sz, oPx, oPz, oP);

    // Pass 2: velocity. Tiles cover zd in [0,999), xd in [0,999).
    dim3 g2((NX - 1 + BX - 1) / BX,   // 16
            (NZ - 1 + BZ - 1) / BZ,   // 63
            NSRC);                    // 16
    k_velocity<<<g2, 256, 0, stream>>>(oP, Ax, Az, sxh, szh, oAx, oAz);
}